// EMDLoss_33861522161767
// MI455X (gfx1250) — compile-verified
//
#include <hip/hip_runtime.h>
#include <math.h>

// ---------------------------------------------------------------------------
// EMD loss via per-batch interior-point QP, one workgroup per batch element.
// Cholesky trailing updates (SYRK) use CDNA5 V_WMMA_F32_16X16X4_F32.
// Wave-synchronous (wave32, in-order DS) diagonal factor/solve phases minimize
// workgroup barriers; shuffle-based reductions; blocked substitutions.
// ---------------------------------------------------------------------------

typedef float v2f __attribute__((ext_vector_type(2)));
typedef float v8f __attribute__((ext_vector_type(8)));

namespace {
constexpr int BATCH = 128;
constexpr int RP = 12, CP = 12;
constexpr int NN = RP * CP;        // 144 QP variables
constexpr int MM = NN + RP + CP;   // 168 constraints
constexpr int NT = NN / 16;        // 9 tiles per dim
constexpr int LDH = NN + 1;        // 145: odd stride -> conflict-free column reads
constexpr int NITER = 25;
constexpr int BLOCK = 128;         // 4 waves of 32
constexpr float L2REG = 1.0e-4f;
constexpr float EPSF  = 1.0e-16f;
constexpr float BIGF  = 1.0e12f;

// shared-memory layout (floats)
constexpr int OFF_H    = 0;
constexpr int OFF_P    = OFF_H + NN * LDH;   // dists p (144)
constexpr int OFF_HV   = OFF_P + NN;         // h (168)
constexpr int OFF_X    = OFF_HV + MM;        // x (144)
constexpr int OFF_S    = OFF_X + NN;         // s (168)
constexpr int OFF_Z    = OFF_S + MM;         // z (168)
constexpr int OFF_D    = OFF_Z + MM;         // d = z/s (168)
constexpr int OFF_RZ   = OFF_D + MM;         // r_z then ds (168)
constexpr int OFF_RC   = OFF_RZ + MM;        // r_cent then dz (168)
constexpr int OFF_TV   = OFF_RC + MM;        // (z*r_z - r_cent)/s (168)
constexpr int OFF_V1   = OFF_TV + MM;        // rhs -> Hi_rhs -> dx (144)
constexpr int OFF_V2   = OFF_V1 + NN;        // ones -> Hi_a (144)
constexpr int OFF_RED  = OFF_V2 + NN;        // cross-wave reduce scratch (4)
constexpr int OFF_RSUM = OFF_RED + 4;        // row sums (12)
constexpr int OFF_CSUM = OFF_RSUM + RP;      // col sums (12)
constexpr int OFF_ETAT = OFF_CSUM + CP;
constexpr int OFF_PHIT = OFF_ETAT + RP;
constexpr int OFF_WT   = OFF_PHIT + RP;
constexpr int OFF_ETAO = OFF_WT + RP;
constexpr int OFF_PHIO = OFF_ETAO + CP;
constexpr int OFF_WO   = OFF_PHIO + CP;
constexpr int SMEM_FLOATS = OFF_WO + CP;     // ~22.4k floats (~90 KB) of 320 KB/WGP
} // namespace

// triangular-pair decode tables for SYRK tile enumeration (max T=8 -> 36 pairs)
__device__ __constant__ unsigned char TRI_I[36] = {
  0,1,1,2,2,2,3,3,3,3,4,4,4,4,4,5,5,5,5,5,5,
  6,6,6,6,6,6,6,7,7,7,7,7,7,7,7};
__device__ __constant__ unsigned char TRI_J[36] = {
  0,0,1,0,1,2,0,1,2,3,0,1,2,3,4,0,1,2,3,4,5,
  0,1,2,3,4,5,6,0,1,2,3,4,5,6,7};

__device__ __forceinline__ float waveReduceSum(float v) {
  #pragma unroll
  for (int m = 16; m >= 1; m >>= 1) v += __shfl_xor(v, m, 32);
  return v;
}
__device__ __forceinline__ float waveReduceMin(float v) {
  #pragma unroll
  for (int m = 16; m >= 1; m >>= 1) v = fminf(v, __shfl_xor(v, m, 32));
  return v;
}

__device__ __forceinline__ float blockReduceSum(float v, float* red, int tid) {
  v = waveReduceSum(v);
  if ((tid & 31) == 0) red[tid >> 5] = v;
  __syncthreads();
  float r = red[0] + red[1] + red[2] + red[3];
  __syncthreads();
  return r;
}
__device__ __forceinline__ float blockReduceMin(float v, float* red, int tid) {
  v = waveReduceMin(v);
  if ((tid & 31) == 0) red[tid >> 5] = v;
  __syncthreads();
  float r = fminf(fminf(red[0], red[1]), fminf(red[2], red[3]));
  __syncthreads();
  return r;
}

__global__ void __launch_bounds__(BLOCK)
emd_ipm_kernel(const float* __restrict__ jet_out,
               const float* __restrict__ jet_target,
               float* __restrict__ ws) {
  extern __shared__ float smem[];
  float* H    = smem + OFF_H;
  float* pv   = smem + OFF_P;
  float* hv   = smem + OFF_HV;
  float* xv   = smem + OFF_X;
  float* sv   = smem + OFF_S;
  float* zv   = smem + OFF_Z;
  float* dv   = smem + OFF_D;
  float* rz   = smem + OFF_RZ;
  float* rc   = smem + OFF_RC;
  float* tv   = smem + OFF_TV;
  float* v1   = smem + OFF_V1;
  float* v2   = smem + OFF_V2;
  float* red  = smem + OFF_RED;
  float* rsum = smem + OFF_RSUM;
  float* csum = smem + OFF_CSUM;
  float* etaT = smem + OFF_ETAT;
  float* phiT = smem + OFF_PHIT;
  float* wT   = smem + OFF_WT;
  float* etaO = smem + OFF_ETAO;
  float* phiO = smem + OFF_PHIO;
  float* wO   = smem + OFF_WO;

  const int tid  = threadIdx.x;
  const int bidx = blockIdx.x;
  const int lane = tid & 31;
  const int wave = tid >> 5;
  const int hlf  = lane >> 4;   // 0/1: half-wave selector (WMMA layouts, RHS select)
  const int l16  = lane & 15;

  // ---- polar conversion: eta, phi, pt for target (rows) and out (cols) ----
  if (tid < RP) {
    const float* q = jet_target + (bidx * RP + tid) * 4;
    float px = q[1], py = q[2], pz = q[3];
    float pt = sqrtf(px * px + py * py + EPSF);
    etaT[tid] = asinhf(pz / (pt + EPSF));
    phiT[tid] = atan2f(py, px);
    wT[tid]   = pt;
  } else if (tid < RP + CP) {
    int j = tid - RP;
    const float* q = jet_out + (bidx * CP + j) * 4;
    float px = q[1], py = q[2], pz = q[3];
    float pt = sqrtf(px * px + py * py + EPSF);
    etaO[j] = asinhf(pz / (pt + EPSF));
    phiO[j] = atan2f(py, px);
    wO[j]   = pt;
  }
  __syncthreads();

  // ---- pairwise distances p, constraint vector h ----
  for (int a = tid; a < NN; a += BLOCK) {
    int i = a / CP, j = a % CP;
    float de = -(etaT[i] - etaO[j]) + EPSF;
    float dp = -(phiT[i] - phiO[j]) + EPSF;
    pv[a] = sqrtf(de * de + dp * dp);
  }
  for (int a = tid; a < MM; a += BLOCK) {
    float v = 0.0f;
    if (a >= NN && a < NN + RP) v = wT[a - NN];
    else if (a >= NN + RP)      v = wO[a - NN - RP];
    hv[a] = v;
  }
  __syncthreads();

  float sumw1 = blockReduceSum(tid < RP ? wT[tid] : 0.0f, red, tid);
  float sumw2 = blockReduceSum(tid < CP ? wO[tid] : 0.0f, red, tid);
  const float bscal = fminf(sumw1, sumw2);

  // ---- IPM initialization ----
  for (int a = tid; a < NN; a += BLOCK) xv[a] = bscal / (float)NN;
  for (int a = tid; a < MM; a += BLOCK) { sv[a] = 1.0f; zv[a] = 1.0f; }
  float yloc = 0.0f;
  __syncthreads();

  for (int it = 0; it < NITER; ++it) {
    // mu = <s,z>/m
    float part = 0.0f;
    for (int a = tid; a < MM; a += BLOCK) part += sv[a] * zv[a];
    float mu = blockReduceSum(part, red, tid) / (float)MM;

    // row/col sums of x -> G x
    if (tid < RP) {
      float acc = 0.0f;
      for (int j = 0; j < CP; ++j) acc += xv[tid * CP + j];
      rsum[tid] = acc;
    } else if (tid < RP + CP) {
      int j = tid - RP; float acc = 0.0f;
      for (int i = 0; i < RP; ++i) acc += xv[i * CP + j];
      csum[j] = acc;
    }
    __syncthreads();

    // r_z = G x + s - h
    for (int a = tid; a < MM; a += BLOCK) {
      float gx;
      if (a < NN)            gx = -xv[a];
      else if (a < NN + RP)  gx = rsum[a - NN];
      else                   gx = csum[a - NN - RP];
      rz[a] = gx + sv[a] - hv[a];
    }

    // r_y = sum(x) - b
    part = 0.0f;
    for (int a = tid; a < NN; a += BLOCK) part += xv[a];
    float ry = blockReduceSum(part, red, tid) - bscal;

    // r_cent, d = z/s, tv = (z*r_z - r_cent)/s
    for (int a = tid; a < MM; a += BLOCK) {
      float sva = sv[a], zva = zv[a];
      float rca = sva * zva - 0.1f * mu;
      rc[a] = rca;
      dv[a] = zva / sva;
      tv[a] = (zva * rz[a] - rca) / sva;
    }
    __syncthreads();

    // rhs = -(r_x + G^T tv);  v2 = ones
    for (int a = tid; a < NN; a += BLOCK) {
      int i1 = a / CP, j1 = a % CP;
      float gtz = -zv[a] + zv[NN + i1] + zv[NN + RP + j1];
      float rx  = L2REG * xv[a] + pv[a] + gtz + yloc;
      float gtt = -tv[a] + tv[NN + i1] + tv[NN + RP + j1];
      v1[a] = -(rx + gtt);
      v2[a] = 1.0f;
    }

    // H = L2*I + G^T diag(d) G (structured); only the lower tile-triangle is
    // ever read (SYRK C-frags, TRSM panels, solves), skip strict-upper tiles.
    for (int idx = tid; idx < NN * NN; idx += BLOCK) {
      int a = idx / NN, c = idx % NN;
      if ((c >> 4) > (a >> 4)) continue;
      int i1 = a / CP, j1 = a % CP;
      int i2 = c / CP, j2 = c % CP;
      float v = 0.0f;
      if (a == c)   v += L2REG + dv[a];
      if (i1 == i2) v += dv[NN + i1];
      if (j1 == j2) v += dv[NN + RP + j1];
      H[a * LDH + c] = v;
    }
    __syncthreads();

    // ---- blocked Cholesky, tile=16, NT=9; SYRK via V_WMMA_F32_16X16X4_F32 ----
    for (int k = 0; k < NT; ++k) {
      float* Hd = &H[(k * 16) * LDH + k * 16];
      // (a) potrf on diagonal tile: wave 0 only, wave-synchronous (in-order DS)
      if (wave == 0) {
        for (int j = 0; j < 16; ++j) {
          if (lane == 0) Hd[j * LDH + j] = sqrtf(fmaxf(Hd[j * LDH + j], 1e-30f));
          __builtin_amdgcn_wave_barrier();
          if (lane > j && lane < 16) Hd[lane * LDH + j] /= Hd[j * LDH + j];
          __builtin_amdgcn_wave_barrier();
          if (lane > j && lane < 16) {
            float ltj = Hd[lane * LDH + j];
            for (int c = j + 1; c <= lane; ++c)
              Hd[lane * LDH + c] -= ltj * Hd[c * LDH + j];
          }
          __builtin_amdgcn_wave_barrier();
        }
      }
      __syncthreads();
      // (b) TRSM: panel rows solve against Lkk^T (row-independent)
      int nrows = (NT - 1 - k) * 16;
      for (int rr = tid; rr < nrows; rr += BLOCK) {
        int grow = (k + 1) * 16 + rr;
        float* Arow = &H[grow * LDH + k * 16];
        for (int j = 0; j < 16; ++j) {
          float acc = Arow[j];
          for (int t = 0; t < j; ++t) acc -= Arow[t] * Hd[j * LDH + t];
          Arow[j] = acc / Hd[j * LDH + j];
        }
      }
      __syncthreads();
      // (c) SYRK trailing update via WMMA (A loaded negated: D = C - L*L^T)
      int T = NT - 1 - k;
      int npairs = (T * (T + 1)) / 2;
      for (int pidx = wave; pidx < npairs; pidx += 4) {
        int ti = k + 1 + (int)TRI_I[pidx];
        int tj = k + 1 + (int)TRI_J[pidx];
        // C/D fragment: VGPR g holds rows M=g (lanes 0-15) and M=g+8 (lanes 16-31)
        int crow = ti * 16 + hlf * 8;
        int ccol = tj * 16 + l16;
        v8f cf;
        #pragma unroll
        for (int g = 0; g < 8; ++g) cf[g] = H[(crow + g) * LDH + ccol];
        #pragma unroll
        for (int kk = 0; kk < 16; kk += 4) {
          int K0 = k * 16 + kk + hlf * 2;
          v2f af, bf;
          af.x = -H[(ti * 16 + l16) * LDH + K0];      // A(16x4): M=l16, K={K0,K0+1}
          af.y = -H[(ti * 16 + l16) * LDH + K0 + 1];
          bf.x =  H[(tj * 16 + l16) * LDH + K0];      // B(4x16)=L_jk^T: N=l16
          bf.y =  H[(tj * 16 + l16) * LDH + K0 + 1];
          cf = __builtin_amdgcn_wmma_f32_16x16x4_f32(
                 false, af, false, bf, (short)0, cf, false, false);
        }
        #pragma unroll
        for (int g = 0; g < 8; ++g) H[(crow + g) * LDH + ccol] = cf[g];
      }
      __syncthreads();
    }

    // ---- blocked cho_solve on v1 (rhs) and v2 (ones) ----
    // forward: L y = b
    for (int I = 0; I < NT; ++I) {
      if (wave == 0) {  // diag block solve: lanes 0-15 -> v1, 16-31 -> v2
        float* vec = hlf ? v2 : v1;
        int row = I * 16 + l16;
        for (int j = 0; j < 16; ++j) {
          int jr = I * 16 + j;
          if (l16 == j) vec[jr] /= H[jr * LDH + jr];
          __builtin_amdgcn_wave_barrier();
          if (l16 > j) vec[row] -= H[row * LDH + jr] * vec[jr];
          __builtin_amdgcn_wave_barrier();
        }
      }
      __syncthreads();
      int base = (I + 1) * 16;
      int rem = NN - base;
      for (int t = tid; t < 2 * rem; t += BLOCK) {
        float* vec = (t < rem) ? v1 : v2;
        int i = base + ((t < rem) ? t : t - rem);
        float acc = vec[i];
        #pragma unroll
        for (int j = 0; j < 16; ++j) acc -= H[i * LDH + I * 16 + j] * vec[I * 16 + j];
        vec[i] = acc;
      }
      __syncthreads();
    }
    // backward: L^T x = y
    for (int I = NT - 1; I >= 0; --I) {
      if (wave == 0) {
        float* vec = hlf ? v2 : v1;
        int row = I * 16 + l16;
        for (int j = 15; j >= 0; --j) {
          int jr = I * 16 + j;
          if (l16 == j) vec[jr] /= H[jr * LDH + jr];
          __builtin_amdgcn_wave_barrier();
          if (l16 < j) vec[row] -= H[jr * LDH + row] * vec[jr];
          __builtin_amdgcn_wave_barrier();
        }
      }
      __syncthreads();
      int top = I * 16;
      for (int t = tid; t < 2 * top; t += BLOCK) {
        float* vec = (t < top) ? v1 : v2;
        int i = (t < top) ? t : t - top;
        float acc = vec[i];
        #pragma unroll
        for (int j = 0; j < 16; ++j) acc -= H[(I * 16 + j) * LDH + i] * vec[I * 16 + j];
        vec[i] = acc;
      }
      __syncthreads();
    }

    // dy, dx
    part = 0.0f;
    for (int a = tid; a < NN; a += BLOCK) part += v1[a];
    float sum1 = blockReduceSum(part, red, tid);
    part = 0.0f;
    for (int a = tid; a < NN; a += BLOCK) part += v2[a];
    float sum2 = blockReduceSum(part, red, tid);
    float dy = (sum1 + ry) / sum2;

    for (int a = tid; a < NN; a += BLOCK) v1[a] -= dy * v2[a];  // v1 = dx
    __syncthreads();

    if (tid < RP) {
      float acc = 0.0f;
      for (int j = 0; j < CP; ++j) acc += v1[tid * CP + j];
      rsum[tid] = acc;
    } else if (tid < RP + CP) {
      int j = tid - RP; float acc = 0.0f;
      for (int i = 0; i < RP; ++i) acc += v1[i * CP + j];
      csum[j] = acc;
    }
    __syncthreads();

    // ds = -r_z - G dx (into rz); dz = (-r_cent - z*ds)/s (into rc)
    for (int a = tid; a < MM; a += BLOCK) {
      float gdx;
      if (a < NN)            gdx = -v1[a];
      else if (a < NN + RP)  gdx = rsum[a - NN];
      else                   gdx = csum[a - NN - RP];
      float dsv = -rz[a] - gdx;
      float dzv = (-rc[a] - zv[a] * dsv) / sv[a];
      rz[a] = dsv;
      rc[a] = dzv;
    }
    __syncthreads();

    // step lengths
    float pms = BIGF, pmz = BIGF;
    for (int a = tid; a < MM; a += BLOCK) {
      float dsv = rz[a], dzv = rc[a];
      if (dsv < 0.0f) pms = fminf(pms, -sv[a] / dsv);
      if (dzv < 0.0f) pmz = fminf(pmz, -zv[a] / dzv);
    }
    float a_s = blockReduceMin(pms, red, tid);
    float a_z = blockReduceMin(pmz, red, tid);
    float alpha = fminf(1.0f, 0.99f * fminf(a_s, a_z));

    // updates
    for (int a = tid; a < NN; a += BLOCK) xv[a] += alpha * v1[a];
    for (int a = tid; a < MM; a += BLOCK) {
      sv[a] += alpha * rz[a];
      zv[a] += alpha * rc[a];
    }
    yloc += alpha * dy;
    __syncthreads();
  }

  // emd_b = <p, x> + |sum w1 - sum w2|
  float part = 0.0f;
  for (int a = tid; a < NN; a += BLOCK) part += pv[a] * xv[a];
  float dotpx = blockReduceSum(part, red, tid);
  if (tid == 0) ws[bidx] = dotpx + fabsf(sumw1 - sumw2);
}

__global__ void emd_reduce_kernel(const float* __restrict__ ws,
                                  float* __restrict__ out) {
  __shared__ float red[BATCH];
  int tid = threadIdx.x;
  red[tid] = ws[tid];
  __syncthreads();
  #pragma unroll
  for (int sft = BATCH / 2; sft > 0; sft >>= 1) {
    if (tid < sft) red[tid] += red[tid + sft];
    __syncthreads();
  }
  if (tid == 0) out[0] = red[0];
}

extern "C" void kernel_launch(void* const* d_in, const int* in_sizes, int n_in,
                              void* d_out, int out_size, void* d_ws, size_t ws_size,
                              hipStream_t stream) {
  (void)in_sizes; (void)n_in; (void)out_size; (void)ws_size;
  const float* jet_out    = (const float*)d_in[0];
  const float* jet_target = (const float*)d_in[1];
  float* ws  = (float*)d_ws;
  float* out = (float*)d_out;

  size_t shmem = (size_t)SMEM_FLOATS * sizeof(float);  // ~90 KB of 320 KB/WGP LDS
  emd_ipm_kernel<<<BATCH, BLOCK, shmem, stream>>>(jet_out, jet_target, ws);
  emd_reduce_kernel<<<1, BATCH, 0, stream>>>(ws, out);
}